// SoftHd_69853348102991
// MI455X (gfx1250) — compile-verified
//
#include <hip/hip_runtime.h>

#define BB   8
#define NN1  2048
#define NN2  2048
#define DD   512
#define TILE 128
#define KC   32
#define NCHUNK (DD / KC)   // 16
#define LSTR 36   // padded LDS row stride (floats): 144B, 16B aligned, conflict-free

typedef __attribute__((ext_vector_type(2))) float v2f;
typedef __attribute__((ext_vector_type(8))) float v8f;

__device__ __forceinline__ float halfwave_min16(float r) {
    r = fminf(r, __shfl_xor(r, 1, 32));
    r = fminf(r, __shfl_xor(r, 2, 32));
    r = fminf(r, __shfl_xor(r, 4, 32));
    r = fminf(r, __shfl_xor(r, 8, 32));
    return r;
}

// Low 32 bits of a flat pointer into LDS == LDS byte offset (shared aperture
// has zero low bits), usable as the VDST lds-address operand of async loads.
__device__ __forceinline__ unsigned lds_off(const void* p) {
    return (unsigned)(uintptr_t)p;
}

// global -> LDS direct DMA, 16B per lane, tracked by ASYNCcnt (no VGPR data path)
__device__ __forceinline__ void async_copy16(unsigned ldsOff, const float* gaddr) {
    asm volatile("global_load_async_to_lds_b128 %0, %1, off"
                 :: "v"(ldsOff), "v"(gaddr) : "memory");
}

// ---------------------------------------------------------------------------
// Kernel 1: per-row sum of squares + init min buffers to +inf bits
// ---------------------------------------------------------------------------
__global__ __launch_bounds__(256)
void softhd_prep(const float* __restrict__ v1, const float* __restrict__ v2,
                 float* __restrict__ sq1, float* __restrict__ sq2,
                 unsigned* __restrict__ colmin, unsigned* __restrict__ rowmin)
{
    const int row  = blockIdx.x * 8 + (threadIdx.x >> 5);
    const int lane = threadIdx.x & 31;
    const int R1   = BB * NN1;

    const float* src;
    float* sq;
    unsigned* mn;
    if (row < R1) {
        src = v1 + (size_t)row * DD; sq = sq1 + row; mn = colmin + row;
    } else {
        int r2 = row - R1;
        src = v2 + (size_t)r2 * DD;  sq = sq2 + r2;  mn = rowmin + r2;
    }

    float s = 0.f;
    #pragma unroll
    for (int c = 0; c < DD; c += 128) {
        float4 x = *(const float4*)(src + c + lane * 4);
        s += x.x * x.x + x.y * x.y + x.z * x.z + x.w * x.w;
    }
    s += __shfl_xor(s, 1, 32);
    s += __shfl_xor(s, 2, 32);
    s += __shfl_xor(s, 4, 32);
    s += __shfl_xor(s, 8, 32);
    s += __shfl_xor(s, 16, 32);
    if (lane == 0) { *sq = s; *mn = 0x7F800000u; }
}

// ---------------------------------------------------------------------------
// Kernel 2: 128x128 distance tile via V_WMMA_F32_16X16X4_F32,
//           async-DMA double-buffered LDS pipeline, fused min-reduce
// ---------------------------------------------------------------------------
__global__ __launch_bounds__(256)
void softhd_tile(const float* __restrict__ v1, const float* __restrict__ v2,
                 const float* __restrict__ sq1, const float* __restrict__ sq2,
                 const int* __restrict__ sz1, const int* __restrict__ sz2,
                 unsigned* __restrict__ rowmin_g, unsigned* __restrict__ colmin_g)
{
    const int b  = blockIdx.z;
    const int j0 = blockIdx.y * TILE;   // rows of v2
    const int i0 = blockIdx.x * TILE;   // rows of v1
    const int s1 = sz1[b];
    const int s2 = sz2[b];
    if (j0 >= s2 || i0 >= s1) return;   // tile fully masked -> contributes nothing

    __shared__ __align__(16) float lA[2][TILE * LSTR];   // v2 tile [j][k], double buffered
    __shared__ __align__(16) float lB[2][TILE * LSTR];   // v1 tile [i][k], double buffered
    __shared__ float    sq2s[TILE];
    __shared__ float    sq1s[TILE];
    __shared__ unsigned rowRed[TILE];
    __shared__ unsigned colRed[TILE];

    const int tid  = threadIdx.x;
    const int lane = tid & 31;
    const int wave = tid >> 5;        // 0..7
    const int wj   = wave >> 1;       // 0..3 : 32 j's per wave
    const int wi   = wave & 1;        // 0..1 : 64 i's per wave
    const int n    = lane & 15;
    const int hl   = lane >> 4;

    if (tid < TILE) {
        sq2s[tid]   = sq2[(size_t)b * NN2 + j0 + tid];
        rowRed[tid] = 0x7F800000u;
    } else {
        int t = tid - TILE;
        sq1s[t]   = sq1[(size_t)b * NN1 + i0 + t];
        colRed[t] = 0x7F800000u;
    }

    const float* gA = v2 + ((size_t)b * NN2 + j0) * DD;
    const float* gB = v1 + ((size_t)b * NN1 + i0) * DD;

    // per-thread DMA lane: row = p*32 + (tid>>3), 16B column (tid&7)*16 within chunk
    const int lr = tid >> 3;          // 0..31
    const int lc = (tid & 7) * 4;     // 0..28 floats
    const float* gA_t = gA + (size_t)lr * DD + lc;
    const float* gB_t = gB + (size_t)lr * DD + lc;
    unsigned ldsA_t[2], ldsB_t[2];
    ldsA_t[0] = lds_off(&lA[0][lr * LSTR + lc]);
    ldsA_t[1] = lds_off(&lA[1][lr * LSTR + lc]);
    ldsB_t[0] = lds_off(&lB[0][lr * LSTR + lc]);
    ldsB_t[1] = lds_off(&lB[1][lr * LSTR + lc]);

    const v8f vzero = {0.f, 0.f, 0.f, 0.f, 0.f, 0.f, 0.f, 0.f};
    v8f acc[2][4];
    #pragma unroll
    for (int tj = 0; tj < 2; ++tj)
        #pragma unroll
        for (int ti = 0; ti < 4; ++ti) acc[tj][ti] = vzero;

    // kick off chunk 0 DMA into buffer 0 (8 async ops per wave)
    #pragma unroll
    for (int p = 0; p < 4; ++p) {
        async_copy16(ldsA_t[0] + p * 32 * LSTR * 4, gA_t + (size_t)p * 32 * DD);
        async_copy16(ldsB_t[0] + p * 32 * LSTR * 4, gB_t + (size_t)p * 32 * DD);
    }

    for (int c = 0; c < NCHUNK; ++c) {
        const int cur = c & 1;
        if (c + 1 < NCHUNK) {
            const int nxt = cur ^ 1;
            const int kn  = (c + 1) * KC;
            #pragma unroll
            for (int p = 0; p < 4; ++p) {
                async_copy16(ldsA_t[nxt] + p * 32 * LSTR * 4, gA_t + (size_t)p * 32 * DD + kn);
                async_copy16(ldsB_t[nxt] + p * 32 * LSTR * 4, gB_t + (size_t)p * 32 * DD + kn);
            }
            // async loads retire in order: <=8 outstanding => chunk c landed
            asm volatile("s_wait_asynccnt 0x8" ::: "memory");
        } else {
            asm volatile("s_wait_asynccnt 0x0" ::: "memory");
        }
        __syncthreads();

        const float* A  = &lA[cur][0];
        const float* Bm = &lB[cur][0];
        #pragma unroll
        for (int k4 = 0; k4 < KC; k4 += 4) {
            const int kk = k4 + 2 * hl;  // lanes 0-15: K={k4,k4+1}; lanes 16-31: K={k4+2,k4+3}
            v2f aF0 = *(const v2f*)(&A [(wj * 32 +      n) * LSTR + kk]);
            v2f aF1 = *(const v2f*)(&A [(wj * 32 + 16 + n) * LSTR + kk]);
            v2f bF0 = *(const v2f*)(&Bm[(wi * 64 +      n) * LSTR + kk]);
            v2f bF1 = *(const v2f*)(&Bm[(wi * 64 + 16 + n) * LSTR + kk]);
            v2f bF2 = *(const v2f*)(&Bm[(wi * 64 + 32 + n) * LSTR + kk]);
            v2f bF3 = *(const v2f*)(&Bm[(wi * 64 + 48 + n) * LSTR + kk]);
            acc[0][0] = __builtin_amdgcn_wmma_f32_16x16x4_f32(false, aF0, false, bF0, (short)0, acc[0][0], false, false);
            acc[0][1] = __builtin_amdgcn_wmma_f32_16x16x4_f32(false, aF0, false, bF1, (short)0, acc[0][1], false, false);
            acc[0][2] = __builtin_amdgcn_wmma_f32_16x16x4_f32(false, aF0, false, bF2, (short)0, acc[0][2], false, false);
            acc[0][3] = __builtin_amdgcn_wmma_f32_16x16x4_f32(false, aF0, false, bF3, (short)0, acc[0][3], false, false);
            acc[1][0] = __builtin_amdgcn_wmma_f32_16x16x4_f32(false, aF1, false, bF0, (short)0, acc[1][0], false, false);
            acc[1][1] = __builtin_amdgcn_wmma_f32_16x16x4_f32(false, aF1, false, bF1, (short)0, acc[1][1], false, false);
            acc[1][2] = __builtin_amdgcn_wmma_f32_16x16x4_f32(false, aF1, false, bF2, (short)0, acc[1][2], false, false);
            acc[1][3] = __builtin_amdgcn_wmma_f32_16x16x4_f32(false, aF1, false, bF3, (short)0, acc[1][3], false, false);
        }
        __syncthreads();   // all waves done reading buf `cur` before it is re-filled
    }

    // ---- fused epilogue: dist = sqrt(max(sq2+sq1-2c,0)), mask -> +inf, min-reduce
    const float INF = __uint_as_float(0x7F800000u);
    float cmin[4] = {INF, INF, INF, INF};

    #pragma unroll
    for (int tj = 0; tj < 2; ++tj) {
        float rmv[8] = {INF, INF, INF, INF, INF, INF, INF, INF};
        #pragma unroll
        for (int ti = 0; ti < 4; ++ti) {
            const int il   = wi * 64 + ti * 16 + n;           // local i
            const bool mi  = (i0 + il) >= s1;
            const float q1 = sq1s[il];
            #pragma unroll
            for (int v = 0; v < 8; ++v) {
                const int jl  = wj * 32 + tj * 16 + v + 8 * hl;   // local j (C: M=v+8*half, N=lane&15)
                const bool mj = (j0 + jl) >= s2;
                float d2   = sq2s[jl] + q1 - 2.0f * acc[tj][ti][v];
                float dist = __builtin_sqrtf(fmaxf(d2, 0.0f));
                if (mi || mj) dist = INF;
                rmv[v]   = fminf(rmv[v], dist);
                cmin[ti] = fminf(cmin[ti], dist);
            }
        }
        #pragma unroll
        for (int v = 0; v < 8; ++v) {           // min over 16 i-lanes (per half)
            float r = halfwave_min16(rmv[v]);
            if (n == 0)
                atomicMin(&rowRed[wj * 32 + tj * 16 + v + 8 * hl], __float_as_uint(r));
        }
    }
    #pragma unroll
    for (int ti = 0; ti < 4; ++ti) {            // min over the two M-halves
        float cv = fminf(cmin[ti], __shfl_xor(cmin[ti], 16, 32));
        if (hl == 0)
            atomicMin(&colRed[wi * 64 + ti * 16 + n], __float_as_uint(cv));
    }
    __syncthreads();

    if (tid < TILE) {
        atomicMin(&rowmin_g[(size_t)b * NN2 + j0 + tid], rowRed[tid]);
    } else {
        int t = tid - TILE;
        atomicMin(&colmin_g[(size_t)b * NN1 + i0 + t], colRed[t]);
    }
}

// ---------------------------------------------------------------------------
// Kernel 3: per-batch sums and final division
// ---------------------------------------------------------------------------
__global__ __launch_bounds__(256)
void softhd_finish(const unsigned* __restrict__ rowmin_g,
                   const unsigned* __restrict__ colmin_g,
                   const int* __restrict__ sz1, const int* __restrict__ sz2,
                   float* __restrict__ out)
{
    __shared__ float red[256];
    const int b  = blockIdx.x;
    const int s1 = sz1[b];
    const int s2 = sz2[b];

    float sum = 0.f;
    for (int j = threadIdx.x; j < s2; j += 256) {
        float x = __uint_as_float(rowmin_g[(size_t)b * NN2 + j]);
        if (x < 3.0e38f) sum += x;
    }
    for (int i = threadIdx.x; i < s1; i += 256) {
        float x = __uint_as_float(colmin_g[(size_t)b * NN1 + i]);
        if (x < 3.0e38f) sum += x;
    }
    red[threadIdx.x] = sum;
    __syncthreads();
    #pragma unroll
    for (int s = 128; s > 0; s >>= 1) {
        if (threadIdx.x < s) red[threadIdx.x] += red[threadIdx.x + s];
        __syncthreads();
    }
    if (threadIdx.x == 0) out[b] = red[0] / (float)(s1 + s2);
}

// ---------------------------------------------------------------------------
extern "C" void kernel_launch(void* const* d_in, const int* in_sizes, int n_in,
                              void* d_out, int out_size, void* d_ws, size_t ws_size,
                              hipStream_t stream)
{
    const float* v1  = (const float*)d_in[0];
    const int*   sz1 = (const int*)  d_in[2];
    const float* v2  = (const float*)d_in[3];
    const int*   sz2 = (const int*)  d_in[5];
    float* out = (float*)d_out;

    // workspace: sq1[B*N1] | sq2[B*N2] | colmin[B*N1] | rowmin[B*N2]  (256 KB)
    float*    sq1    = (float*)d_ws;
    float*    sq2    = sq1 + (size_t)BB * NN1;
    unsigned* colmin = (unsigned*)(sq2 + (size_t)BB * NN2);
    unsigned* rowmin = colmin + (size_t)BB * NN1;

    softhd_prep<<<dim3((BB * (NN1 + NN2)) / 8), 256, 0, stream>>>(
        v1, v2, sq1, sq2, colmin, rowmin);

    softhd_tile<<<dim3(NN1 / TILE, NN2 / TILE, BB), 256, 0, stream>>>(
        v1, v2, sq1, sq2, sz1, sz2, rowmin, colmin);

    softhd_finish<<<dim3(BB), 256, 0, stream>>>(rowmin, colmin, sz1, sz2, out);
}